// SparseMLP_24910810317383
// MI455X (gfx1250) — compile-verified
//
#include <hip/hip_runtime.h>
#include <cstdint>

// Problem constants (match reference: B=4096, N=8192, K=0.1, K_PERCENT=0.5)
constexpr int      BROWS  = 4096;
constexpr int      NCOLS  = 8192;
constexpr int      KROW   = 820;        // ceil(0.1 * 8192)
constexpr uint32_t KBATCH = 1679360u;   // ceil(0.5 * 4096 * 820)
constexpr int      GWIN   = 1024;       // per-block LDS window of hi-16 bins

// Monotone key: larger float  <->  larger unsigned key
__device__ __forceinline__ uint32_t fkey(uint32_t f) {
  return f ^ (0x80000000u | (uint32_t)((int32_t)f >> 31));
}

// CDNA5 async global->LDS 16B copy (tracked by ASYNCcnt).
// GV addressing mode: LDS address taken from the vdst VGPR; global addr in VGPR pair.
__device__ __forceinline__ void async_copy_b128(const float* g, uint32_t* lds) {
  uint32_t lofs = (uint32_t)(uintptr_t)lds;   // low 32 bits = LDS byte address
  uint64_t gptr = (uint64_t)(uintptr_t)g;
  asm volatile("global_load_async_to_lds_b128 %0, %1, off"
               :: "v"(lofs), "v"(gptr)
               : "memory");
}

__device__ __forceinline__ void wait_async_zero() {
  asm volatile("s_wait_asynccnt 0x0" ::: "memory");
}

// ---------------------------------------------------------------------------
// K0: zero the global histograms + selector block
// ---------------------------------------------------------------------------
__global__ void zero_kernel(uint32_t* __restrict__ p, int n) {
  int i = blockIdx.x * blockDim.x + threadIdx.x;
  if (i < n) p[i] = 0u;
}

// ---------------------------------------------------------------------------
// K1: per-row exact 820th-largest key via 4-pass radix select in LDS,
//     plus windowed accumulation of survivor hi-16 bins into global hist.
// ---------------------------------------------------------------------------
__global__ __launch_bounds__(256) void row_select_kernel(
    const float* __restrict__ x, uint32_t* __restrict__ row_key,
    uint32_t* __restrict__ gh_hi) {
  __shared__ uint32_t srow[NCOLS];   // 32 KB row staging (raw float bits)
  __shared__ uint32_t hist[256];
  __shared__ uint32_t ctrl[2];
  __shared__ uint32_t gwin[GWIN];

  const int t   = threadIdx.x;
  const int row = blockIdx.x;
  const float* rp = x + (size_t)row * NCOLS;

  // Stage row into LDS with async copies: 2048 b128 chunks.
#pragma unroll
  for (int j = 0; j < 8; ++j) {
    int c = t + 256 * j;
    async_copy_b128(rp + c * 4, &srow[c * 4]);
  }
  wait_async_zero();
  __syncthreads();

  // Cache 32 monotone keys per thread in VGPRs (bank-conflict-free reads).
  uint32_t uk[32];
#pragma unroll
  for (int j = 0; j < 32; ++j) uk[j] = fkey(srow[j * 256 + t]);

  // 4-pass MSB-first radix select for the KROW-th largest key.
  uint32_t prefix = 0u, pmask = 0u, krem = KROW;
  for (int pass = 0; pass < 4; ++pass) {
    const int shift = 24 - 8 * pass;
    hist[t] = 0u;
    __syncthreads();
#pragma unroll
    for (int j = 0; j < 32; ++j) {
      uint32_t u = uk[j];
      if ((u & pmask) == prefix) atomicAdd(&hist[(u >> shift) & 0xFFu], 1u);
    }
    __syncthreads();
    if (t == 0) {
      uint32_t cum = 0u; int b = 255;
      for (; b > 0; --b) { uint32_t c = hist[b]; if (cum + c >= krem) break; cum += c; }
      ctrl[0] = prefix | ((uint32_t)b << shift);
      ctrl[1] = krem - cum;
    }
    __syncthreads();
    prefix = ctrl[0];
    krem   = ctrl[1];
    pmask |= (0xFFu << shift);
  }

  const uint32_t rk = prefix;              // exact key of row's 820th largest
  if (t == 0) row_key[row] = rk;

  // Survivor hi-16 histogram: accumulate in an LDS window anchored at the
  // row threshold bin, flush only nonzero bins (few hundred global atomics).
  for (int i = t; i < GWIN; i += 256) gwin[i] = 0u;
  __syncthreads();
  const uint32_t base = rk >> 16;
#pragma unroll
  for (int j = 0; j < 32; ++j) {
    uint32_t u = uk[j];
    if (u >= rk) {
      uint32_t off = (u >> 16) - base;     // u>=rk  =>  (u>>16) >= base
      if (off < (uint32_t)GWIN) atomicAdd(&gwin[off], 1u);
      else                      atomicAdd(&gh_hi[u >> 16], 1u);  // rare
    }
  }
  __syncthreads();
  for (int i = t; i < GWIN; i += 256) {
    uint32_t c = gwin[i];
    if (c) atomicAdd(&gh_hi[base + i], c);
  }
}

// ---------------------------------------------------------------------------
// K2: scan 64K hi-bins top-down -> bin containing the KBATCH-th survivor.
//     sel[0]=hi_bin, sel[1]=remaining count inside that bin.
// ---------------------------------------------------------------------------
__global__ __launch_bounds__(256) void scan_hi_kernel(
    const uint32_t* __restrict__ gh, uint32_t* __restrict__ sel) {
  __shared__ uint32_t part[256];
  const int t = threadIdx.x;
  uint32_t s = 0u;
  const uint32_t* p = gh + t * 256;
  for (int i = 0; i < 256; ++i) s += p[i];
  part[t] = s;
  __syncthreads();
  if (t == 0) {
    uint32_t k = KBATCH, cum = 0u; int ch = 255;
    for (; ch > 0; --ch) { uint32_t c = part[ch]; if (cum + c >= k) break; cum += c; }
    uint32_t rem = k - cum;
    const uint32_t* q = gh + ch * 256;
    uint32_t cum2 = 0u; int b = 255;
    for (; b > 0; --b) { uint32_t c = q[b]; if (cum2 + c >= rem) break; cum2 += c; }
    sel[0] = (uint32_t)(ch * 256 + b);
    sel[1] = rem - cum2;
  }
}

// ---------------------------------------------------------------------------
// K3: streaming pass (mostly L2 hits): survivors in the chosen hi-bin
//     histogram their low 16 key bits.
// ---------------------------------------------------------------------------
__global__ __launch_bounds__(256) void lo_hist_kernel(
    const float4* __restrict__ x, const uint32_t* __restrict__ row_key,
    const uint32_t* __restrict__ sel, uint32_t* __restrict__ gl) {
  size_t q = (size_t)blockIdx.x * blockDim.x + threadIdx.x;   // float4 index
  const uint32_t binhi = sel[0];
  const uint32_t rk = row_key[q >> 11];                       // 2048 float4/row
  float4 v = x[q];
  uint32_t u;
  u = fkey(__float_as_uint(v.x)); if (u >= rk && (u >> 16) == binhi) atomicAdd(&gl[u & 0xFFFFu], 1u);
  u = fkey(__float_as_uint(v.y)); if (u >= rk && (u >> 16) == binhi) atomicAdd(&gl[u & 0xFFFFu], 1u);
  u = fkey(__float_as_uint(v.z)); if (u >= rk && (u >> 16) == binhi) atomicAdd(&gl[u & 0xFFFFu], 1u);
  u = fkey(__float_as_uint(v.w)); if (u >= rk && (u >> 16) == binhi) atomicAdd(&gl[u & 0xFFFFu], 1u);
}

// ---------------------------------------------------------------------------
// K4: scan 64K lo-bins -> exact 32-bit global threshold key. sel[2]=g_key.
// ---------------------------------------------------------------------------
__global__ __launch_bounds__(256) void scan_lo_kernel(
    const uint32_t* __restrict__ gl, uint32_t* __restrict__ sel) {
  __shared__ uint32_t part[256];
  const int t = threadIdx.x;
  uint32_t s = 0u;
  const uint32_t* p = gl + t * 256;
  for (int i = 0; i < 256; ++i) s += p[i];
  part[t] = s;
  __syncthreads();
  if (t == 0) {
    uint32_t k = sel[1], binhi = sel[0], cum = 0u; int ch = 255;
    for (; ch > 0; --ch) { uint32_t c = part[ch]; if (cum + c >= k) break; cum += c; }
    uint32_t rem = k - cum;
    const uint32_t* q = gl + ch * 256;
    uint32_t cum2 = 0u; int b = 255;
    for (; b > 0; --b) { uint32_t c = q[b]; if (cum2 + c >= rem) break; cum2 += c; }
    sel[2] = (binhi << 16) | (uint32_t)(ch * 256 + b);
  }
}

// ---------------------------------------------------------------------------
// K5: streaming masked copy: keep iff key >= max(row_key, g_key).
// ---------------------------------------------------------------------------
__global__ __launch_bounds__(256) void finalize_kernel(
    const float4* __restrict__ x, const uint32_t* __restrict__ row_key,
    const uint32_t* __restrict__ sel, float4* __restrict__ out) {
  size_t q = (size_t)blockIdx.x * blockDim.x + threadIdx.x;
  const uint32_t gk = sel[2];
  uint32_t rk = row_key[q >> 11];
  const uint32_t th = rk > gk ? rk : gk;
  float4 v = x[q];
  v.x = (fkey(__float_as_uint(v.x)) >= th) ? v.x : 0.0f;
  v.y = (fkey(__float_as_uint(v.y)) >= th) ? v.y : 0.0f;
  v.z = (fkey(__float_as_uint(v.z)) >= th) ? v.z : 0.0f;
  v.w = (fkey(__float_as_uint(v.w)) >= th) ? v.w : 0.0f;
  out[q] = v;
}

// ---------------------------------------------------------------------------
extern "C" void kernel_launch(void* const* d_in, const int* in_sizes, int n_in,
                              void* d_out, int out_size, void* d_ws, size_t ws_size,
                              hipStream_t stream) {
  (void)in_sizes; (void)n_in; (void)out_size; (void)ws_size;
  const float* x = (const float*)d_in[0];
  float* out = (float*)d_out;

  uint32_t* row_key = (uint32_t*)d_ws;        // 4096
  uint32_t* gh_hi   = row_key + BROWS;        // 65536
  uint32_t* gh_lo   = gh_hi + 65536;          // 65536
  uint32_t* sel     = gh_lo + 65536;          // 4 (contiguous after gh_lo)

  const int nzero = 65536 * 2 + 4;            // gh_hi + gh_lo + sel
  const int nvec4 = (BROWS * NCOLS) / 4;      // 8,388,608 float4
  const int vblocks = nvec4 / 256;            // 32768

  zero_kernel<<<(nzero + 255) / 256, 256, 0, stream>>>(gh_hi, nzero);
  row_select_kernel<<<BROWS, 256, 0, stream>>>(x, row_key, gh_hi);
  scan_hi_kernel<<<1, 256, 0, stream>>>(gh_hi, sel);
  lo_hist_kernel<<<vblocks, 256, 0, stream>>>((const float4*)x, row_key, sel, gh_lo);
  scan_lo_kernel<<<1, 256, 0, stream>>>(gh_lo, sel);
  finalize_kernel<<<vblocks, 256, 0, stream>>>((const float4*)x, row_key, sel, (float4*)out);
}